// DistRelConv_37967510897365
// MI455X (gfx1250) — compile-verified
//
#include <hip/hip_runtime.h>

typedef __attribute__((ext_vector_type(2))) float v2f;
typedef __attribute__((ext_vector_type(8))) float v8f;

static constexpr int FD   = 64;   // in_feats == out_feats
static constexpr int KD   = 128;  // 2*FD  (GEMM K)
static constexpr int NUMS = 4;    // sampled intermediates per edge
static constexpr int TILE = 16;   // edges per WMMA M-tile
static constexpr int BLK  = 128;  // 4 wave32 per block
static constexpr int CST  = 132;  // padded cat row stride (floats) -> conflict-free ds_load_b64

// ---------------------------------------------------------------------------
// Zero output + degree arrays (must re-init every call: atomic accumulation)
// ---------------------------------------------------------------------------
__global__ void init_kernel(float* __restrict__ out, float* __restrict__ in_deg,
                            float* __restrict__ out_deg, int n_out, int n_nodes) {
    int stride = gridDim.x * blockDim.x;
    int gid = blockIdx.x * blockDim.x + threadIdx.x;
    for (int i = gid; i < n_out; i += stride) out[i] = 0.0f;
    for (int i = gid; i < n_nodes; i += stride) { in_deg[i] = 0.0f; out_deg[i] = 0.0f; }
}

// ---------------------------------------------------------------------------
// Degree histogram via f32 atomics (L2-resident, 50K*2 floats)
// ---------------------------------------------------------------------------
__global__ void degree_kernel(const int* __restrict__ src, const int* __restrict__ dst,
                              float* __restrict__ in_deg, float* __restrict__ out_deg, int E) {
    int i = blockIdx.x * blockDim.x + threadIdx.x;
    if (i < E) {
        atomicAdd(&in_deg[dst[i]], 1.0f);
        atomicAdd(&out_deg[src[i]], 1.0f);
    }
}

// ---------------------------------------------------------------------------
// EG = embed_table @ G_w.T   ([NB,32] x [32,64] -> [NB,64]) — tiny, done once.
// Collapses the per-edge embedding->G projection into a table lookup.
// ---------------------------------------------------------------------------
__global__ void eg_kernel(const float* __restrict__ embed, const float* __restrict__ G_w,
                          float* __restrict__ EG, int nb, int dd) {
    int idx = blockIdx.x * blockDim.x + threadIdx.x;
    if (idx < nb * FD) {
        int i = idx >> 6;       // embedding row
        int j = idx & 63;       // output feature
        float s = 0.0f;
        for (int k = 0; k < dd; ++k) s += embed[i * dd + k] * G_w[j * dd + k];
        EG[idx] = s;
    }
}

// ---------------------------------------------------------------------------
// Main fused edge kernel.
//   Per 16-edge tile:  scalars -> cat[16x128] in LDS -> 32x v_wmma_f32_16x16x4_f32
//   per wave (wave nt owns output cols [16nt,16nt+16), B-frags stationary in VGPRs)
//   -> scale/bias -> global_atomic_add_f32 scatter to out[dst].
// ---------------------------------------------------------------------------
__global__ __launch_bounds__(BLK) void edge_wmma_kernel(
    const float* __restrict__ feat, const float* __restrict__ loc,
    const float* __restrict__ bounds, int nb1,
    const float* __restrict__ EG,        // [NB][64] (precomputed)
    const float* __restrict__ agg_w,     // [64][128] row-major (out x in)
    const float* __restrict__ agg_b,     // [64]
    const int* __restrict__ src, const int* __restrict__ dst,
    const int* __restrict__ inter,       // [E][4]
    const float* __restrict__ in_deg, const float* __restrict__ out_deg,
    float* __restrict__ out, int E)
{
    __shared__ __align__(16) float s_eg[64 * FD];       // 16 KB: EG table resident in LDS
    __shared__ __align__(16) float s_cat[TILE * CST];   // 8.25 KB padded cat tile
    __shared__ float s_bounds[64];
    __shared__ float s_scale[TILE];
    __shared__ int   s_sidx[TILE], s_didx[TILE], s_b1[TILE];
    __shared__ int   s_iidx[TILE * NUMS], s_bk[TILE * NUMS];

    const int tid  = threadIdx.x;
    const int lane = tid & 31;
    const int wave = tid >> 5;

    // ---- block preamble: stage EG + boundaries into LDS ----
    {
        const float4* gsrc = (const float4*)EG;
        float4* gdst = (float4*)s_eg;
        for (int i = tid; i < 64 * FD / 4; i += BLK) gdst[i] = gsrc[i];
    }
    for (int i = tid; i < 64; i += BLK)
        s_bounds[i] = (i < nb1) ? bounds[i] : 3.4e38f;

    // ---- stationary B fragments: wave `wave` owns N-cols [16*wave, 16*wave+16) ----
    // V_WMMA_F32_16X16X4_F32 B (4x16): VGPR0 = rows {k0, k0+2} (lanes 0-15 / 16-31),
    // VGPR1 = rows {k0+1, k0+3}.  B[k][n] = agg_w[n*128 + k] -> k-consecutive = b64 load.
    const int ncol = (wave << 4) + (lane & 15);
    const int koff = (lane >> 4) << 1;
    v2f Bf[32];
#pragma unroll
    for (int ks = 0; ks < 32; ++ks)
        Bf[ks] = *(const v2f*)(agg_w + ncol * KD + (ks << 2) + koff);
    const float bias = agg_b[ncol];

    __syncthreads();

    const int tiles = (E + TILE - 1) / TILE;
    for (int t = blockIdx.x; t < tiles; t += gridDim.x) {
        const int ebase = t * TILE;
        __syncthreads();  // previous tile's LDS readers done before overwrite

        // ---- phase 1: per-edge scalars (16 threads) ----
        if (tid < TILE) {
            const int e = ebase + tid;
            if (e < E) {
                const int s = src[e], d = dst[e];
                s_sidx[tid] = s; s_didx[tid] = d;
                const float sx = loc[s * 3 + 0], sy = loc[s * 3 + 1], sz = loc[s * 3 + 2];
                {
                    const float dx = loc[d * 3 + 0] - sx, dy = loc[d * 3 + 1] - sy,
                                dz = loc[d * 3 + 2] - sz;
                    const float dist1 = sqrtf(dx * dx + dy * dy + dz * dz);
                    int lo = 0, hi = nb1;                 // searchsorted(side='left')
                    while (lo < hi) { int mid = (lo + hi) >> 1;
                                      if (s_bounds[mid] < dist1) lo = mid + 1; else hi = mid; }
                    s_b1[tid] = lo;
                }
                for (int k = 0; k < NUMS; ++k) {
                    const int ii = inter[e * NUMS + k];
                    s_iidx[tid * NUMS + k] = ii;
                    const float dx = sx - loc[ii * 3 + 0], dy = sy - loc[ii * 3 + 1],
                                dz = sz - loc[ii * 3 + 2];
                    const float dk = sqrtf(dx * dx + dy * dy + dz * dz);
                    int lo = 0, hi = nb1;
                    while (lo < hi) { int mid = (lo + hi) >> 1;
                                      if (s_bounds[mid] < dk) lo = mid + 1; else hi = mid; }
                    s_bk[tid * NUMS + k] = lo;
                }
                float di = in_deg[d];  if (di < 1.0f) di = 1.0f;   // clip(deg,1)^-0.5
                float dg = out_deg[s]; if (dg < 1.0f) dg = 1.0f;
                s_scale[tid] = (1.0f / sqrtf(di)) * (1.0f / sqrtf(dg));
            } else {
                s_sidx[tid] = 0; s_didx[tid] = 0; s_b1[tid] = 0; s_scale[tid] = 0.0f;
                for (int k = 0; k < NUMS; ++k) { s_iidx[tid * NUMS + k] = 0; s_bk[tid * NUMS + k] = 0; }
            }
        }
        __syncthreads();

        // ---- phase 2: build cat[16][128] (8 threads per edge, 16 cols each) ----
        {
            const int el = tid >> 3;
            const int part = tid & 7;
            const int fbase = (part & 3) << 4;   // 0,16,32,48
            float vals[16];
            if (part < 4) {
                // cat[:, 0:64] = EG[b1] * feat[src]   (coalesced float4 row gathers)
                const int b1 = s_b1[el];
                const float4* fr = (const float4*)(feat + (size_t)s_sidx[el] * FD + fbase);
                const float4* er = (const float4*)(s_eg + b1 * FD + fbase);
#pragma unroll
                for (int c = 0; c < 4; ++c) {
                    const float4 fv = fr[c]; const float4 ev = er[c];
                    vals[c * 4 + 0] = fv.x * ev.x; vals[c * 4 + 1] = fv.y * ev.y;
                    vals[c * 4 + 2] = fv.z * ev.z; vals[c * 4 + 3] = fv.w * ev.w;
                }
                float* crow = s_cat + el * CST + fbase;
#pragma unroll
                for (int j = 0; j < 16; ++j) crow[j] = vals[j];
            } else {
                // cat[:, 64:128] = mean_k( EG[bk] * feat[inter_k] )
#pragma unroll
                for (int j = 0; j < 16; ++j) vals[j] = 0.0f;
                for (int k = 0; k < NUMS; ++k) {
                    const int bk = s_bk[el * NUMS + k];
                    const float4* fr = (const float4*)(feat + (size_t)s_iidx[el * NUMS + k] * FD + fbase);
                    const float4* er = (const float4*)(s_eg + bk * FD + fbase);
#pragma unroll
                    for (int c = 0; c < 4; ++c) {
                        const float4 fv = fr[c]; const float4 ev = er[c];
                        vals[c * 4 + 0] += fv.x * ev.x; vals[c * 4 + 1] += fv.y * ev.y;
                        vals[c * 4 + 2] += fv.z * ev.z; vals[c * 4 + 3] += fv.w * ev.w;
                    }
                }
                float* crow = s_cat + el * CST + FD + fbase;
#pragma unroll
                for (int j = 0; j < 16; ++j) crow[j] = vals[j] * 0.25f;
            }
        }
        __syncthreads();

        // ---- phase 3: 32 K-steps of fp32 WMMA; A-frag from padded LDS cat ----
        // A (16x4): lane<16 -> (M=lane, K=k0..k0+1); lane>=16 -> (M=lane-16, K=k0+2..k0+3)
        v8f acc = {};
        const float* arow = s_cat + (lane & 15) * CST + koff;
#pragma unroll
        for (int ks = 0; ks < 32; ++ks) {
            const v2f a = *(const v2f*)(arow + (ks << 2));
            acc = __builtin_amdgcn_wmma_f32_16x16x4_f32(
                /*neg_a=*/false, a, /*neg_b=*/false, Bf[ks],
                /*c_mod=*/(short)0, acc, /*reuse_a=*/false, /*reuse_b=*/false);
        }

        // ---- phase 4: msg = scale*(acc+bias), atomic scatter to out[dst] ----
        // C/D layout: VGPR v, lanes 0-15 -> M=v; lanes 16-31 -> M=v+8; N = lane&15
        const int rbase = (lane >> 4) << 3;
#pragma unroll
        for (int v = 0; v < 8; ++v) {
            const int m = rbase + v;
            const float val = s_scale[m] * (acc[v] + bias);
            atomicAdd(out + (size_t)s_didx[m] * FD + ncol, val);
        }
    }
}

// ---------------------------------------------------------------------------
extern "C" void kernel_launch(void* const* d_in, const int* in_sizes, int n_in,
                              void* d_out, int out_size, void* d_ws, size_t ws_size,
                              hipStream_t stream) {
    const float* feat   = (const float*)d_in[0];
    const float* loc    = (const float*)d_in[1];
    const float* bounds = (const float*)d_in[2];
    const float* embed  = (const float*)d_in[3];
    const float* G_w    = (const float*)d_in[4];
    const float* agg_w  = (const float*)d_in[5];
    const float* agg_b  = (const float*)d_in[6];
    const int*   src    = (const int*)d_in[7];
    const int*   dst    = (const int*)d_in[8];
    const int*   inter  = (const int*)d_in[9];
    float* out = (float*)d_out;

    const int N   = in_sizes[0] / FD;       // 50000
    const int E   = in_sizes[7];            // 500000
    const int nb1 = in_sizes[2];            // 63 boundaries
    const int dd  = in_sizes[4] / FD;       // dist_dim = 32
    const int nb  = in_sizes[3] / dd;       // 64 embedding rows

    // workspace: [in_deg N][out_deg N][EG nb*64]
    float* in_deg  = (float*)d_ws;
    float* out_deg = in_deg + N;
    float* EG      = out_deg + N;

    const int n_out = N * FD;
    init_kernel<<<4096, 256, 0, stream>>>(out, in_deg, out_deg, n_out, N);
    degree_kernel<<<(E + 255) / 256, 256, 0, stream>>>(src, dst, in_deg, out_deg, E);
    eg_kernel<<<(nb * FD + 255) / 256, 256, 0, stream>>>(embed, G_w, EG, nb, dd);

    const int tiles = (E + TILE - 1) / TILE;
    int grid = tiles < 2048 ? tiles : 2048;
    edge_wmma_kernel<<<grid, BLK, 0, stream>>>(feat, loc, bounds, nb1, EG, agg_w, agg_b,
                                               src, dst, inter, in_deg, out_deg, out, E);
}